// RWKVBlock_52956946760510
// MI455X (gfx1250) — compile-verified
//
#include <hip/hip_runtime.h>
#include <math.h>
#include <stdint.h>

// ---------------- problem constants ----------------
#define T_DIM 1024
#define C_DIM 2048
#define H_DIM 32
#define N_DIM 64
#define DLORA 64
#define DG 128

typedef float v2f __attribute__((ext_vector_type(2)));
typedef float v8f __attribute__((ext_vector_type(8)));

__device__ __forceinline__ float sigmoidf_(float x) { return 1.f / (1.f + __expf(-x)); }

// Async copy of 16 bytes global -> LDS per active lane (ASYNCcnt-tracked).
// GV addressing: full 64-bit global address in a VGPR pair, LDS byte offset
// in a VGPR (low 32 bits of the generic shared pointer).
__device__ __forceinline__ void async_g2l_b128(void* lds_ptr, const void* gptr)
{
    uint32_t lo = (uint32_t)(uintptr_t)lds_ptr;
    uint64_t ga = (uint64_t)(uintptr_t)gptr;
    asm volatile("global_load_async_to_lds_b128 %0, %1, off"
                 :: "v"(lo), "v"(ga) : "memory");
}

// =====================================================================
// Tiled f32 WMMA GEMM:  out[T x N] (=|+=) epi( A'[T x K] @ W[K x N] )
//   A' = A                          (a_mixed == 0)
//   A' = A + (shift(A) - A) * mix   (a_mixed == 1, token-shift fusion;
//                                    row -1 comes from prev0[])
// Block = 128 threads (4 wave32), computes a 64x64 tile.
// - Weights stream via GLOBAL_LOAD_ASYNC_TO_LDS_B128 into a double-
//   buffered LDS tile; next tile's HBM fetch overlaps current WMMAs.
//   Per-wave in-order ASYNCcnt: s_wait_asynccnt 4 retires the older tile.
// - A tile staged through VALU (needed for the fused token-shift mix).
// - V_WMMA_F32_16X16X4_F32: the problem is HBM-bound on f32 weights
//   (~200 MB -> ~8.6 us floor at 23.3 TB/s), so exact f32 WMMA with zero
//   conversion traffic is the right matrix unit.
// =====================================================================
#define GTM 64
#define GTN 64
#define GTK 32
#define LSTR 68   // LDS row stride in floats (272 B: 16B-aligned rows, de-phased banks)

__global__ __launch_bounds__(128)
void gemm_wmma_f32(const float* __restrict__ A, const float* __restrict__ prev0,
                   const float* __restrict__ mix, const float* __restrict__ W,
                   float* __restrict__ out, int Trows, int K, int N,
                   int a_mixed, int epi, int addTo)
{
    __shared__ float As[GTK][LSTR];
    __shared__ float Bs[2][GTK][LSTR];
    const int tid  = threadIdx.x;
    const int wave = tid >> 5;
    const int lane = tid & 31;
    const int half = lane >> 4;      // 0: lanes 0-15, 1: lanes 16-31
    const int lm   = lane & 15;
    const int tBase = blockIdx.y * GTM;
    const int nBase = blockIdx.x * GTN;

    v8f acc[4] = {};

    // ---- async-stage one 32x64 B tile (8 KB) into Bs[buf]: 4 issues x 128 lanes x 16 B
    auto issueB = [&](int kB, int buf) {
        #pragma unroll
        for (int i = 0; i < 4; ++i) {
            int f4 = tid + i * 128;          // 0..511 float4 slots
            int kl = f4 >> 4;                // row in tile (16 float4 per row)
            int n4 = f4 & 15;
            async_g2l_b128(&Bs[buf][kl][n4 * 4],
                           &W[(size_t)(kB + kl) * N + nBase + n4 * 4]);
        }
    };

    issueB(0, 0);
    int buf = 0;

    for (int kB = 0; kB < K; kB += GTK, buf ^= 1) {
        // ---- stage A tile (64 rows x 32 k) through VALU with fused mix ----
        #pragma unroll
        for (int i = 0; i < 16; ++i) {
            int idx = tid + i * 128;          // 0..2047
            int m  = idx >> 5;
            int kl = idx & 31;
            int t  = tBase + m;
            int k  = kB + kl;
            float v = A[(size_t)t * K + k];
            if (a_mixed) {
                float p = (t == 0) ? prev0[k] : A[(size_t)(t - 1) * K + k];
                v = v + (p - v) * mix[k];
            }
            As[kl][m] = v;
        }

        // ---- pipeline: kick off next weight tile, retire current one ----
        if (kB + GTK < K) {
            issueB(kB + GTK, buf ^ 1);
            asm volatile("s_wait_asynccnt 0x4" ::: "memory");  // older 4 done (in-order)
        } else {
            asm volatile("s_wait_asynccnt 0x0" ::: "memory");
        }
        __syncthreads();

        // ---- 8 WMMA K-steps of 4, over 4 N-subtiles each ----
        #pragma unroll
        for (int ks = 0; ks < GTK; ks += 4) {
            v2f a;
            a.x = As[ks + 2 * half + 0][16 * wave + lm];
            a.y = As[ks + 2 * half + 1][16 * wave + lm];
            #pragma unroll
            for (int nt = 0; nt < 4; ++nt) {
                v2f b;
                b.x = Bs[buf][ks + 2 * half + 0][nt * 16 + lm];
                b.y = Bs[buf][ks + 2 * half + 1][nt * 16 + lm];
                acc[nt] = __builtin_amdgcn_wmma_f32_16x16x4_f32(
                    false, a, false, b, (short)0, acc[nt], false, false);
            }
        }
        __syncthreads();
    }

    // ---- epilogue: C/D layout -> vgpr i holds M=i (lanes 0-15) / M=i+8 ----
    #pragma unroll
    for (int nt = 0; nt < 4; ++nt) {
        #pragma unroll
        for (int i = 0; i < 8; ++i) {
            int row = tBase + 16 * wave + i + 8 * half;
            int col = nBase + nt * 16 + lm;
            float v = acc[nt][i];
            if (epi == 1)      v = tanhf(v);
            else if (epi == 2) v = sigmoidf_(v);
            else if (epi == 3) { v = v > 0.f ? v : 0.f; v = v * v; }
            size_t o = (size_t)row * N + col;
            if (addTo) out[o] += v; else out[o] = v;
        }
    }
}

// =====================================================================
// LayerNorm over one row of length Cn (block = 256 threads per row)
// =====================================================================
__global__ __launch_bounds__(256)
void ln_kernel(const float* __restrict__ in, const float* __restrict__ w,
               const float* __restrict__ b, float* __restrict__ out, int Cn)
{
    __shared__ float red[256];
    const int t = blockIdx.x, tid = threadIdx.x;
    const float* row = in + (size_t)t * Cn;
    float s = 0.f;
    for (int c = tid; c < Cn; c += 256) s += row[c];
    red[tid] = s; __syncthreads();
    for (int o = 128; o > 0; o >>= 1) { if (tid < o) red[tid] += red[tid + o]; __syncthreads(); }
    float mean = red[0] / Cn; __syncthreads();
    float vs = 0.f;
    for (int c = tid; c < Cn; c += 256) { float d = row[c] - mean; vs += d * d; }
    red[tid] = vs; __syncthreads();
    for (int o = 128; o > 0; o >>= 1) { if (tid < o) red[tid] += red[tid + o]; __syncthreads(); }
    float rstd = rsqrtf(red[0] / Cn + 1e-5f);
    for (int c = tid; c < Cn; c += 256)
        out[(size_t)t * Cn + c] = (row[c] - mean) * rstd * w[c] + b[c];
}

// =====================================================================
// Pre-scan fusion, one block per (head, t), 64 threads:
//   a = sigmoid(a0 + a_pre);  kk = L2norm_head(k*k_k);
//   k <- k*(1+(a-1)*k_a);  v <- v + (v_first - v)*sigmoid(v0 + vg_pre);
//   w <- exp(-0.606531 * sigmoid(w0 + w_pre))
// In-place over the GEMM output buffers.
// =====================================================================
__global__ __launch_bounds__(64)
void fuse_pre_scan(const float* __restrict__ k_k, const float* __restrict__ k_a,
                   const float* __restrict__ a0, const float* __restrict__ v0,
                   const float* __restrict__ w0, const float* __restrict__ v_first,
                   float* __restrict__ kb, float* __restrict__ vb,
                   float* __restrict__ wb, float* __restrict__ ab,
                   const float* __restrict__ vgb, float* __restrict__ kkb)
{
    __shared__ float red[64];
    const int h = blockIdx.x, t = blockIdx.y, tid = threadIdx.x;
    const int c = h * N_DIM + tid;
    const size_t idx = (size_t)t * C_DIM + c;

    float a    = sigmoidf_(a0[c] + ab[idx]);
    float kraw = kb[idx];
    float kkv  = kraw * k_k[c];
    red[tid] = kkv * kkv; __syncthreads();
    for (int o = 32; o > 0; o >>= 1) { if (tid < o) red[tid] += red[tid + o]; __syncthreads(); }
    float nrm = fmaxf(sqrtf(red[0]), 1e-12f);

    kkb[idx] = kkv / nrm;
    kb[idx]  = kraw * (1.f + (a - 1.f) * k_a[c]);
    ab[idx]  = a;
    float vg   = sigmoidf_(v0[c] + vgb[idx]);
    float vraw = vb[idx];
    vb[idx]  = vraw + (v_first[idx] - vraw) * vg;
    wb[idx]  = __expf(-0.606531f * sigmoidf_(w0[c] + wb[idx]));
}

// =====================================================================
// Sequential delta-rule scan. One block per head (64 threads = 2 wave32);
// thread i owns state row i: 64 f32 live in VGPRs for the whole scan.
//   sab = s . kk ;  s = s*w - sab*(kk*a) + v_i*k ;  out_i = s . r
// =====================================================================
__global__ __launch_bounds__(64)
void wkv_scan(const float* __restrict__ state0, const float* __restrict__ wb,
              const float* __restrict__ kkb, const float* __restrict__ ab,
              const float* __restrict__ kb, const float* __restrict__ vb,
              const float* __restrict__ rb, float* __restrict__ outs)
{
    const int h = blockIdx.x, i = threadIdx.x;
    __shared__ float Lw[64], Lkk[64], La[64], Lk[64], Lv[64], Lr[64];

    float s[N_DIM];
    #pragma unroll
    for (int j = 0; j < N_DIM; ++j)
        s[j] = state0[(size_t)h * N_DIM * N_DIM + (size_t)i * N_DIM + j];

    for (int t = 0; t < T_DIM; ++t) {
        const size_t base = (size_t)t * C_DIM + h * N_DIM;
        Lw[i] = wb[base + i]; Lkk[i] = kkb[base + i]; La[i] = ab[base + i];
        Lk[i] = kb[base + i]; Lv[i]  = vb[base + i];  Lr[i] = rb[base + i];
        __syncthreads();

        float sab = 0.f;
        #pragma unroll
        for (int j = 0; j < N_DIM; ++j) sab += s[j] * Lkk[j];
        const float vi = Lv[i];
        float o = 0.f;
        #pragma unroll
        for (int j = 0; j < N_DIM; ++j) {
            float sj = s[j] * Lw[j] - sab * (Lkk[j] * La[j]) + vi * Lk[j];
            s[j] = sj;
            o += sj * Lr[j];
        }
        outs[base + i] = o;
        __syncthreads();
    }
}

// =====================================================================
// Post-scan: per-(t,head) groupnorm (eps 6.4e-4) + bonus + gate.
//   outs <- (gn(outs)*lnx_w+lnx_b + (sum r*k*r_k)*v) * g     (in place)
// =====================================================================
__global__ __launch_bounds__(64)
void post_scan(const float* __restrict__ lnx_w, const float* __restrict__ lnx_b,
               const float* __restrict__ r_k, const float* __restrict__ rb,
               const float* __restrict__ kb, const float* __restrict__ vb,
               const float* __restrict__ gb, float* __restrict__ outs)
{
    __shared__ float r0[64], r1[64], r2[64];
    const int h = blockIdx.x, t = blockIdx.y, tid = threadIdx.x;
    const int c = h * N_DIM + tid;
    const size_t idx = (size_t)t * C_DIM + c;

    float o  = outs[idx];
    float rk = rb[idx] * kb[idx] * r_k[c];
    r0[tid] = o; r1[tid] = o * o; r2[tid] = rk;
    __syncthreads();
    for (int off = 32; off > 0; off >>= 1) {
        if (tid < off) { r0[tid] += r0[tid + off]; r1[tid] += r1[tid + off]; r2[tid] += r2[tid + off]; }
        __syncthreads();
    }
    float mean = r0[0] * (1.f / 64.f);
    float var  = r1[0] * (1.f / 64.f) - mean * mean;
    float gn   = (o - mean) * rsqrtf(var + 0.00064f) * lnx_w[c] + lnx_b[c];
    float bonus = r2[0] * vb[idx];
    outs[idx] = (gn + bonus) * gb[idx];
}

// =====================================================================
// Host-side orchestration
// =====================================================================
extern "C" void kernel_launch(void* const* d_in, const int* in_sizes, int n_in,
                              void* d_out, int out_size, void* d_ws, size_t ws_size,
                              hipStream_t stream)
{
    (void)in_sizes; (void)n_in; (void)out_size; (void)ws_size;
    const float* x          = (const float*)d_in[0];
    const float* x_prev_att = (const float*)d_in[1];
    const float* vk_state   = (const float*)d_in[2];
    const float* x_prev_ffn = (const float*)d_in[3];
    const float* v_first    = (const float*)d_in[4];
    const float* ln1_w = (const float*)d_in[5],  *ln1_b = (const float*)d_in[6];
    const float* x_r = (const float*)d_in[7],  *x_w = (const float*)d_in[8];
    const float* x_k = (const float*)d_in[9],  *x_v = (const float*)d_in[10];
    const float* x_a = (const float*)d_in[11], *x_g = (const float*)d_in[12];
    const float* w0 = (const float*)d_in[13], *w1 = (const float*)d_in[14], *w2 = (const float*)d_in[15];
    const float* a0 = (const float*)d_in[16], *a1 = (const float*)d_in[17], *a2 = (const float*)d_in[18];
    const float* v0 = (const float*)d_in[19], *v1 = (const float*)d_in[20], *v2 = (const float*)d_in[21];
    const float* g1 = (const float*)d_in[22], *g2 = (const float*)d_in[23];
    const float* k_k = (const float*)d_in[24], *k_a = (const float*)d_in[25], *r_k = (const float*)d_in[26];
    const float* R_ = (const float*)d_in[27], *K_ = (const float*)d_in[28];
    const float* V_ = (const float*)d_in[29], *O_ = (const float*)d_in[30];
    const float* lnx_w = (const float*)d_in[31], *lnx_b = (const float*)d_in[32];
    const float* ln2_w = (const float*)d_in[33], *ln2_b = (const float*)d_in[34];
    const float* ffn_x_k = (const float*)d_in[35];
    const float* ffn_K = (const float*)d_in[36], *ffn_V = (const float*)d_in[37];
    float* out = (float*)d_out;

    // ---- workspace carve-up (f32) ----
    const size_t TC = (size_t)T_DIM * C_DIM;
    float* ws   = (float*)d_ws;
    float* xln  = ws;             // T*C  (reused for ln2 output)
    float* rb   = ws + 1 * TC;    // r
    float* kb   = ws + 2 * TC;    // k  -> modified k
    float* vb   = ws + 3 * TC;    // v  -> blended v
    float* wb   = ws + 4 * TC;    // w_pre -> decay w
    float* ab   = ws + 5 * TC;    // a_pre -> a
    float* vgb  = ws + 6 * TC;    // v-gate pre-sigmoid
    float* kkb  = ws + 7 * TC;    // normalized kk
    float* gb   = ws + 8 * TC;    // g
    float* outs = ws + 9 * TC;    // scan output -> gated attn input
    float* kf   = ws + 10 * TC;   // FFN hidden (T x 4C)
    float* wh1  = ws + 14 * TC;                    // T x 64
    float* ah1  = wh1 + (size_t)T_DIM * DLORA;     // T x 64
    float* vh1  = ah1 + (size_t)T_DIM * DLORA;     // T x 64
    float* gh1  = vh1 + (size_t)T_DIM * DLORA;     // T x 128

    const dim3 blk128(128), blk256(256), blk64(64);
    const dim3 gCC(C_DIM / GTN, T_DIM / GTM);       // 32 x 16
    const dim3 gC64(DLORA / GTN, T_DIM / GTM);      // 1 x 16
    const dim3 gC128(DG / GTN, T_DIM / GTM);        // 2 x 16
    const dim3 gFFN(4 * C_DIM / GTN, T_DIM / GTM);  // 128 x 16
    const dim3 gHT(H_DIM, T_DIM);

    // ---- TMix ----
    ln_kernel<<<T_DIM, blk256, 0, stream>>>(x, ln1_w, ln1_b, xln, C_DIM);

    gemm_wmma_f32<<<gCC, blk128, 0, stream>>>(xln, x_prev_att, x_r, R_, rb,  T_DIM, C_DIM, C_DIM, 1, 0, 0);
    gemm_wmma_f32<<<gCC, blk128, 0, stream>>>(xln, x_prev_att, x_k, K_, kb,  T_DIM, C_DIM, C_DIM, 1, 0, 0);
    gemm_wmma_f32<<<gCC, blk128, 0, stream>>>(xln, x_prev_att, x_v, V_, vb,  T_DIM, C_DIM, C_DIM, 1, 0, 0);

    gemm_wmma_f32<<<gC64, blk128, 0, stream>>>(xln, x_prev_att, x_w, w1, wh1, T_DIM, C_DIM, DLORA, 1, 1, 0); // tanh
    gemm_wmma_f32<<<gCC,  blk128, 0, stream>>>(wh1, nullptr, nullptr, w2, wb,  T_DIM, DLORA, C_DIM, 0, 0, 0);
    gemm_wmma_f32<<<gC64, blk128, 0, stream>>>(xln, x_prev_att, x_a, a1, ah1, T_DIM, C_DIM, DLORA, 1, 0, 0);
    gemm_wmma_f32<<<gCC,  blk128, 0, stream>>>(ah1, nullptr, nullptr, a2, ab,  T_DIM, DLORA, C_DIM, 0, 0, 0);
    gemm_wmma_f32<<<gC64, blk128, 0, stream>>>(xln, x_prev_att, x_v, v1, vh1, T_DIM, C_DIM, DLORA, 1, 0, 0);
    gemm_wmma_f32<<<gCC,  blk128, 0, stream>>>(vh1, nullptr, nullptr, v2, vgb, T_DIM, DLORA, C_DIM, 0, 0, 0);
    gemm_wmma_f32<<<gC128, blk128, 0, stream>>>(xln, x_prev_att, x_g, g1, gh1, T_DIM, C_DIM, DG, 1, 2, 0);   // sigmoid
    gemm_wmma_f32<<<gCC,   blk128, 0, stream>>>(gh1, nullptr, nullptr, g2, gb,  T_DIM, DG, C_DIM, 0, 0, 0);

    fuse_pre_scan<<<gHT, blk64, 0, stream>>>(k_k, k_a, a0, v0, w0, v_first,
                                             kb, vb, wb, ab, vgb, kkb);
    wkv_scan<<<H_DIM, blk64, 0, stream>>>(vk_state, wb, kkb, ab, kb, vb, rb, outs);
    post_scan<<<gHT, blk64, 0, stream>>>(lnx_w, lnx_b, r_k, rb, kb, vb, gb, outs);

    // x = x + outs @ O_
    hipMemcpyAsync(out, x, TC * sizeof(float), hipMemcpyDeviceToDevice, stream);
    gemm_wmma_f32<<<gCC, blk128, 0, stream>>>(outs, nullptr, nullptr, O_, out, T_DIM, C_DIM, C_DIM, 0, 0, 1);

    // ---- CMix ----
    ln_kernel<<<T_DIM, blk256, 0, stream>>>(out, ln2_w, ln2_b, xln, C_DIM);
    gemm_wmma_f32<<<gFFN, blk128, 0, stream>>>(xln, x_prev_ffn, ffn_x_k, ffn_K, kf,
                                               T_DIM, C_DIM, 4 * C_DIM, 1, 3, 0);   // relu^2
    gemm_wmma_f32<<<gCC, blk128, 0, stream>>>(kf, nullptr, nullptr, ffn_V, out,
                                              T_DIM, 4 * C_DIM, C_DIM, 0, 0, 1);
}